// simclr_11991548690656
// MI455X (gfx1250) — compile-verified
//
#include <hip/hip_runtime.h>
#include <hip/hip_bf16.h>

#define N_NODES 100000
#define N_EDGES 1600000
#define DIM     64
#define N_LAYERS 5
#define N_GRAPHS 1024
#define EMB     (DIM * N_LAYERS)   // 320
#define BN_EPS  1e-5f
#define NSTRIPS (N_NODES / 16)     // 6250 (exact)
#define WPAD    65                 // padded LDS row stride (floats)

typedef __attribute__((ext_vector_type(2))) float v2f;
typedef __attribute__((ext_vector_type(8))) float v8f;

// ---------------- utility kernels ----------------

__global__ void zero_f32(float* __restrict__ p, int n) {
    int i = blockIdx.x * blockDim.x + threadIdx.x;
    if (i < n) p[i] = 0.0f;
}

__global__ void copy_f32v4(const float4* __restrict__ a, float4* __restrict__ b, int n4) {
    int i = blockIdx.x * blockDim.x + threadIdx.x;
    if (i < n4) b[i] = a[i];
}

__global__ void graph_counts(const int* __restrict__ batch, float* __restrict__ counts, int n) {
    int i = blockIdx.x * blockDim.x + threadIdx.x;
    if (i < n) atomicAdd(&counts[batch[i]], 1.0f);
}

// ---------------- edge scatter: z[dst] += w * h[src] ----------------
__global__ void edge_scatter(const int* __restrict__ src, const int* __restrict__ dst,
                             const float* __restrict__ ew, const float* __restrict__ h,
                             float* __restrict__ z) {
    int tid = blockIdx.x * blockDim.x + threadIdx.x;     // < N_EDGES*16
    if (tid >= N_EDGES * 16) return;
    int e  = tid >> 4;
    int f0 = (tid & 15) << 2;
    int s = src[e], d = dst[e];
    float w = ew[e];
    const float4 hv = *reinterpret_cast<const float4*>(h + (size_t)s * DIM + f0);
    float* zp = z + (size_t)d * DIM + f0;
    atomicAdd(zp + 0, w * hv.x);
    atomicAdd(zp + 1, w * hv.y);
    atomicAdd(zp + 2, w * hv.z);
    atomicAdd(zp + 3, w * hv.w);
}

// ---------------- fused GIN MLP layer ----------------
// H = ReLU( ReLU(Z @ W1 + b1) @ W2 + b2 ), plus per-feature sum/sumsq into stats.
// One wave32 computes one 16-row strip (all 64 output features) with
// V_WMMA_F32_16X16X4_F32. Weights live transposed in LDS; the intermediate
// 16x64 tile stays in a per-wave LDS region (never touches global memory).
__global__ __launch_bounds__(256) void gin_mlp_fused(
    const float* __restrict__ Z,  const float* __restrict__ W1,
    const float* __restrict__ b1, const float* __restrict__ W2,
    const float* __restrict__ b2, float* __restrict__ H,
    float* __restrict__ stats)
{
    __shared__ float w1t[64 * WPAD];          // W1^T, padded
    __shared__ float w2t[64 * WPAD];          // W2^T, padded
    __shared__ float tbuf[8 * 16 * WPAD];     // per-wave 16x64 intermediate
    __shared__ float sstat[128];              // block stats: [sum | sumsq]

    // stage transposed weights (coalesced global read, conflict-free LDS write)
    for (int i = threadIdx.x; i < 4096; i += 256) {
        int r = i >> 6, c = i & 63;
        w1t[c * WPAD + r] = W1[i];
        w2t[c * WPAD + r] = W2[i];
    }
    if (threadIdx.x < 128) sstat[threadIdx.x] = 0.0f;
    __syncthreads();

    const int wave  = threadIdx.x >> 5;
    const int lane  = threadIdx.x & 31;
    const int strip = blockIdx.x * 8 + wave;
    const int col16 = lane & 15;
    const int khalf = (lane >> 4) << 1;       // 0 or 2
    const int r0    = (lane >> 4) << 3;       // C-layout row base: 0 or 8
    float* tw = tbuf + wave * 16 * WPAD;

    if (strip < NSTRIPS) {                    // wave-uniform: EXEC all-ones inside
        const int row0 = strip << 4;

        // A fragments for GEMM1: z[row0+col16][k..k+1], all 16 K-steps in regs
        v2f afr[16];
        const float* zp = Z + (size_t)(row0 + col16) * DIM + khalf;
        #pragma unroll
        for (int kk = 0; kk < 16; ++kk) {
            float2 av = *reinterpret_cast<const float2*>(zp + kk * 4);
            afr[kk].x = av.x; afr[kk].y = av.y;
        }

        // GEMM1: t = ReLU(z @ W1 + b1) -> per-wave LDS tile
        #pragma unroll
        for (int nt = 0; nt < 4; ++nt) {
            v8f acc = {0.f,0.f,0.f,0.f,0.f,0.f,0.f,0.f};
            const float* wcol = w1t + (nt * 16 + col16) * WPAD + khalf;
            #pragma unroll
            for (int kk = 0; kk < 16; ++kk) {
                v2f b; b.x = wcol[kk * 4]; b.y = wcol[kk * 4 + 1];
                acc = __builtin_amdgcn_wmma_f32_16x16x4_f32(
                          false, afr[kk], false, b, (short)0, acc, false, false);
            }
            float bv = b1[nt * 16 + col16];
            #pragma unroll
            for (int r = 0; r < 8; ++r)
                tw[(r0 + r) * WPAD + nt * 16 + col16] = fmaxf(acc[r] + bv, 0.0f);
        }

        // A fragments for GEMM2 from the LDS tile (same-wave DS ordering)
        #pragma unroll
        for (int kk = 0; kk < 16; ++kk) {
            afr[kk].x = tw[col16 * WPAD + kk * 4 + khalf];
            afr[kk].y = tw[col16 * WPAD + kk * 4 + khalf + 1];
        }

        // GEMM2: h = ReLU(t @ W2 + b2) -> global, + wave-level stats into LDS
        #pragma unroll
        for (int nt = 0; nt < 4; ++nt) {
            v8f acc = {0.f,0.f,0.f,0.f,0.f,0.f,0.f,0.f};
            const float* wcol = w2t + (nt * 16 + col16) * WPAD + khalf;
            #pragma unroll
            for (int kk = 0; kk < 16; ++kk) {
                v2f b; b.x = wcol[kk * 4]; b.y = wcol[kk * 4 + 1];
                acc = __builtin_amdgcn_wmma_f32_16x16x4_f32(
                          false, afr[kk], false, b, (short)0, acc, false, false);
            }
            float bv = b2[nt * 16 + col16];
            float s1 = 0.f, s2 = 0.f;
            #pragma unroll
            for (int r = 0; r < 8; ++r) {
                float v = fmaxf(acc[r] + bv, 0.0f);
                H[(size_t)(row0 + r0 + r) * DIM + nt * 16 + col16] = v;
                s1 += v; s2 += v * v;
            }
            atomicAdd(&sstat[nt * 16 + col16], s1);         // ds_add_f32
            atomicAdd(&sstat[64 + nt * 16 + col16], s2);
        }
    }
    __syncthreads();
    if (threadIdx.x < 128) atomicAdd(&stats[threadIdx.x], sstat[threadIdx.x]);
}

// ---------------- generic fp32 WMMA GEMM (projection head, K=320) ----------------
__global__ void wmma_gemm_f32(const float* __restrict__ A, const float* __restrict__ B,
                              const float* __restrict__ bias, float* __restrict__ C,
                              int M, int K, int N, int relu) {
    int gwave = (blockIdx.x * blockDim.x + threadIdx.x) >> 5;
    int lane  = threadIdx.x & 31;
    int tilesN = N >> 4;
    int totalTiles = (M >> 4) * tilesN;
    if (gwave >= totalTiles) return;           // wave-uniform
    int mt = gwave / tilesN;
    int nt = gwave % tilesN;

    int arow  = (mt << 4) + (lane & 15);
    int bcol  = (nt << 4) + (lane & 15);
    int khalf = (lane >> 4) << 1;

    v8f acc = {0.f,0.f,0.f,0.f,0.f,0.f,0.f,0.f};
    const float* Ap = A + (size_t)arow * K + khalf;
    #pragma unroll 4
    for (int k = 0; k < K; k += 4) {
        v2f a, b;
        a.x = Ap[k];
        a.y = Ap[k + 1];
        b.x = B[(size_t)(k + khalf)     * N + bcol];
        b.y = B[(size_t)(k + khalf + 1) * N + bcol];
        acc = __builtin_amdgcn_wmma_f32_16x16x4_f32(
                  false, a, false, b, (short)0, acc, false, false);
    }

    float bv = bias[bcol];
    int crow0 = (mt << 4) + ((lane >> 4) << 3);
    #pragma unroll
    for (int r = 0; r < 8; ++r) {
        float v = acc[r] + bv;
        if (relu) v = fmaxf(v, 0.0f);
        C[(size_t)(crow0 + r) * N + bcol] = v;
    }
}

// ---------------- batchnorm apply (in-place) + pooled accumulation ----------------
__global__ void bn_pool(float* __restrict__ h, const float* __restrict__ stats,
                        const float* __restrict__ gamma, const float* __restrict__ beta,
                        const int* __restrict__ batch, float* __restrict__ pooled,
                        int layer) {
    int tid = blockIdx.x * blockDim.x + threadIdx.x;      // < N_NODES*64
    if (tid >= N_NODES * DIM) return;
    int node = tid >> 6;
    int f    = tid & 63;
    const float inv_n = 1.0f / (float)N_NODES;
    float mean = stats[f] * inv_n;
    float var  = stats[DIM + f] * inv_n - mean * mean;
    float sc   = gamma[f] * rsqrtf(var + BN_EPS);
    float sh   = beta[f] - mean * sc;
    float v = h[tid] * sc + sh;
    h[tid] = v;
    atomicAdd(&pooled[(size_t)batch[node] * EMB + layer * DIM + f], v);
}

__global__ void pool_div(float* __restrict__ pooled, const float* __restrict__ counts) {
    int i = blockIdx.x * blockDim.x + threadIdx.x;        // < N_GRAPHS*EMB
    if (i >= N_GRAPHS * EMB) return;
    float c = fmaxf(counts[i / EMB], 1.0f);
    pooled[i] = pooled[i] / c;
}

// ---------------- host orchestration ----------------

extern "C" void kernel_launch(void* const* d_in, const int* in_sizes, int n_in,
                              void* d_out, int out_size, void* d_ws, size_t ws_size,
                              hipStream_t stream) {
    (void)in_sizes; (void)n_in; (void)out_size; (void)ws_size;

    const float* x    = (const float*)d_in[0];
    const int*   ei   = (const int*)  d_in[1];   // [2, N_EDGES]
    const float* ew   = (const float*)d_in[2];
    const int*   batch= (const int*)  d_in[3];
    const float* W1s  = (const float*)d_in[4];   // [L,64,64]
    const float* b1s  = (const float*)d_in[5];
    const float* W2s  = (const float*)d_in[6];
    const float* b2s  = (const float*)d_in[7];
    const float* gam  = (const float*)d_in[8];
    const float* bet  = (const float*)d_in[9];
    const float* pW1  = (const float*)d_in[10];  // [320,320]
    const float* pb1  = (const float*)d_in[11];
    const float* pW2  = (const float*)d_in[12];
    const float* pb2  = (const float*)d_in[13];

    const int* esrc = ei;
    const int* edst = ei + N_EDGES;

    const size_t ND = (size_t)N_NODES * DIM;     // 6,400,000 floats
    float* ws = (float*)d_ws;
    size_t off = 0;
    float* h      = ws + off; off += ND;
    float* z      = ws + off; off += ND;
    float* stats  = ws + off; off += 128;
    float* pooled = ws + off; off += (size_t)N_GRAPHS * EMB;
    float* counts = ws + off; off += N_GRAPHS;
    float* t1     = ws + off; off += (size_t)N_GRAPHS * EMB;

    const int TB = 256;
    const int nd4 = (int)(ND / 4);

    // h = x
    copy_f32v4<<<(nd4 + TB - 1) / TB, TB, 0, stream>>>((const float4*)x, (float4*)h, nd4);
    // zero pooled + counts
    zero_f32<<<(N_GRAPHS * EMB + TB - 1) / TB, TB, 0, stream>>>(pooled, N_GRAPHS * EMB);
    zero_f32<<<(N_GRAPHS + TB - 1) / TB, TB, 0, stream>>>(counts, N_GRAPHS);
    // per-graph node counts
    graph_counts<<<(N_NODES + TB - 1) / TB, TB, 0, stream>>>(batch, counts, N_NODES);

    const int fusedBlocks = (NSTRIPS + 7) / 8;   // 8 strips (waves) per block

    for (int l = 0; l < N_LAYERS; ++l) {
        // z = h
        copy_f32v4<<<(nd4 + TB - 1) / TB, TB, 0, stream>>>((const float4*)h, (float4*)z, nd4);
        // zero stats
        zero_f32<<<1, 128, 0, stream>>>(stats, 128);
        // z[dst] += w * h[src]
        edge_scatter<<<(N_EDGES * 16 + TB - 1) / TB, TB, 0, stream>>>(esrc, edst, ew, h, z);
        // h = ReLU(ReLU(z@W1+b1)@W2+b2), stats accumulated
        gin_mlp_fused<<<fusedBlocks, TB, 0, stream>>>(z,
                                                      W1s + (size_t)l * DIM * DIM,
                                                      b1s + l * DIM,
                                                      W2s + (size_t)l * DIM * DIM,
                                                      b2s + l * DIM,
                                                      h, stats);
        // BN apply in-place + pooled accumulation
        bn_pool<<<(N_NODES * DIM + TB - 1) / TB, TB, 0, stream>>>(h, stats,
                                                                  gam + l * DIM, bet + l * DIM,
                                                                  batch, pooled, l);
    }

    // pooled /= counts
    pool_div<<<(N_GRAPHS * EMB + TB - 1) / TB, TB, 0, stream>>>(pooled, counts);

    // projection head: t1 = ReLU(pooled @ pW1 + pb1); out = t1 @ pW2 + pb2
    const int projTiles = (N_GRAPHS / 16) * (EMB / 16);   // 1280
    const int projBlocks = (projTiles + 7) / 8;
    wmma_gemm_f32<<<projBlocks, TB, 0, stream>>>(pooled, pW1, pb1, t1,
                                                 N_GRAPHS, EMB, EMB, 1);
    wmma_gemm_f32<<<projBlocks, TB, 0, stream>>>(t1, pW2, pb2, (float*)d_out,
                                                 N_GRAPHS, EMB, EMB, 0);
}